// WrapperModel2_22462678958524
// MI455X (gfx1250) — compile-verified
//
#include <hip/hip_runtime.h>
#include <stdint.h>

#define N_ANCH   33600
#define NC       80
#define NCH      84
#define TILE_W   128
#define N_PAD    65536          // next pow2 >= N_ANCH
#define TOPN     30000
#define CONF_T   0.25f
#define MAX_WH_F 7680.0f

typedef unsigned int uint32x4_t __attribute__((ext_vector_type(4)));
typedef int          int32x4_t  __attribute__((ext_vector_type(4)));
typedef int          int32x8_t  __attribute__((ext_vector_type(8)));

// ---------------------------------------------------------------------------
// Stage 1: TDM-stage an 84 x 128 channel-major tile into LDS, then per-anchor
// class max/argmax + xywh->xyxy; emit raw rows and 64-bit sort keys.
// key = orderedBits(conf or -inf) << 32 | ~anchor_index  (descending sort,
// ties resolved toward smaller index; padding keys are 0 and sort last).
// ---------------------------------------------------------------------------
__global__ void stage1_kernel(const float* __restrict__ pred,
                              float* __restrict__ rows_raw,
                              unsigned long long* __restrict__ keys)
{
    __shared__ float tile[NCH * TILE_W];   // 42 KB of the 320 KB WGP LDS
    const unsigned blk = blockIdx.x;
    const unsigned col = threadIdx.x;
    const unsigned i   = blk * TILE_W + col;
    const unsigned rem = N_ANCH - blk * TILE_W;   // grid sized so rem > 0

    if (threadIdx.x == 0) {
        unsigned long long ga = (unsigned long long)(const void*)(pred + (size_t)blk * TILE_W);
        unsigned           la = (unsigned)(unsigned long long)(void*)&tile[0];
        // D# group0: count=1 | lds_addr | global_addr | type=2
        uint32x4_t g0;
        g0[0] = 1u;
        g0[1] = la;
        g0[2] = (unsigned)ga;
        g0[3] = ((unsigned)(ga >> 32) & 0x01FFFFFFu) | (2u << 30);
        // D# group1: data_size=4B; tensor_dim0=rem, tensor_dim1=84;
        // tile_dim0=128, tile_dim1=84; tensor_dim0_stride=33600
        int32x8_t g1;
        g1[0] = (int)(2u << 16);                              // wg_mask=0, data_size=4B
        g1[1] = (int)((rem & 0xFFFFu) << 16);                 // tensor_dim0 lo
        g1[2] = (int)((rem >> 16) | ((unsigned)NCH << 16));   // tensor_dim0 hi | tensor_dim1 lo
        g1[3] = (int)((unsigned)TILE_W << 16);                // tensor_dim1 hi=0 | tile_dim0
        g1[4] = (int)NCH;                                     // tile_dim1 | tile_dim2=0
        g1[5] = (int)N_ANCH;                                  // tensor_dim0_stride lo
        g1[6] = 0;                                            // stride hi | dim1_stride lo
        g1[7] = 0;
        int32x4_t gz = {0, 0, 0, 0};
#if __clang_major__ >= 23
        int32x8_t gz8 = {0, 0, 0, 0, 0, 0, 0, 0};
        __builtin_amdgcn_tensor_load_to_lds(g0, g1, gz, gz, gz8, 0);
#else
        __builtin_amdgcn_tensor_load_to_lds(g0, g1, gz, gz, 0);
#endif
    }
    __builtin_amdgcn_s_wait_tensorcnt(0);   // waves 1..3 have TENSORcnt==0, pass through
    __syncthreads();

    if (i < N_ANCH) {
        const float xc = tile[0 * TILE_W + col];
        const float yc = tile[1 * TILE_W + col];
        const float w  = tile[2 * TILE_W + col];
        const float h  = tile[3 * TILE_W + col];
        float best = tile[4 * TILE_W + col];
        int   bj   = 0;
#pragma unroll 4
        for (int c = 1; c < NC; ++c) {
            float v = tile[(4 + c) * TILE_W + col];
            if (v > best) { best = v; bj = c; }   // strict > keeps first occurrence
        }
        const float hw = w * 0.5f, hh = h * 0.5f;
        rows_raw[(size_t)i * 6 + 0] = xc - hw;
        rows_raw[(size_t)i * 6 + 1] = yc - hh;
        rows_raw[(size_t)i * 6 + 2] = xc + hw;
        rows_raw[(size_t)i * 6 + 3] = yc + hh;
        rows_raw[(size_t)i * 6 + 4] = best;
        rows_raw[(size_t)i * 6 + 5] = (float)bj;

        const float sk = (best > CONF_T) ? best : -__builtin_inff();
        unsigned u = __float_as_uint(sk);
        u = (u & 0x80000000u) ? ~u : (u | 0x80000000u);   // total-order map
        keys[i] = ((unsigned long long)u << 32) | (unsigned)(~i);
    }
}

__global__ void pad_keys_kernel(unsigned long long* __restrict__ keys)
{
    unsigned i = N_ANCH + blockIdx.x * blockDim.x + threadIdx.x;
    if (i < N_PAD) keys[i] = 0ull;   // sorts strictly last (descending)
}

// ---------------------------------------------------------------------------
// Bitonic sort, descending, 65536 x u64. Global passes for j >= 2048,
// all j <= 1024 passes of a stage fused in LDS (1024 thr / 2048 elem / 16 KB).
// ---------------------------------------------------------------------------
__global__ void bitonic_global_kernel(unsigned long long* __restrict__ keys,
                                      unsigned k, unsigned j)
{
    unsigned tid = blockIdx.x * blockDim.x + threadIdx.x;      // 0..32767
    unsigned i   = ((tid & ~(j - 1u)) << 1) | (tid & (j - 1u));
    unsigned ixj = i | j;
    bool desc = ((i & k) == 0u);
    unsigned long long a = keys[i], b = keys[ixj];
    if ((a < b) == desc) { keys[i] = b; keys[ixj] = a; }
}

__global__ void bitonic_lds_kernel(unsigned long long* __restrict__ keys,
                                   unsigned k, unsigned jstart)
{
    __shared__ unsigned long long sh[2048];
    const unsigned base = blockIdx.x * 2048u;
    sh[threadIdx.x]          = keys[base + threadIdx.x];
    sh[threadIdx.x + 1024u]  = keys[base + threadIdx.x + 1024u];
    __syncthreads();
    for (unsigned j = jstart; j >= 1u; j >>= 1) {
        unsigned tid = threadIdx.x;
        unsigned i   = ((tid & ~(j - 1u)) << 1) | (tid & (j - 1u));
        unsigned ixj = i | j;
        bool desc = (((base + i) & k) == 0u);
        unsigned long long a = sh[i], b = sh[ixj];
        if ((a < b) == desc) { sh[i] = b; sh[ixj] = a; }
        __syncthreads();
    }
    keys[base + threadIdx.x]         = sh[threadIdx.x];
    keys[base + threadIdx.x + 1024u] = sh[threadIdx.x + 1024u];
}

// ---------------------------------------------------------------------------
// Stage 3: top-30000 gather -> rows / boxes / scores (concatenated flat).
// ---------------------------------------------------------------------------
__global__ void gather_out_kernel(const unsigned long long* __restrict__ keys,
                                  const float* __restrict__ rows_raw,
                                  float* __restrict__ out)
{
    unsigned i = blockIdx.x * blockDim.x + threadIdx.x;
    if (i >= TOPN) return;
    unsigned idx = ~(unsigned)(keys[i] & 0xFFFFFFFFull);
    float r0 = 0.f, r1 = 0.f, r2 = 0.f, r3 = 0.f, r4 = 0.f, r5 = 0.f;
    if (idx < N_ANCH) {
        r0 = rows_raw[(size_t)idx * 6 + 0];
        r1 = rows_raw[(size_t)idx * 6 + 1];
        r2 = rows_raw[(size_t)idx * 6 + 2];
        r3 = rows_raw[(size_t)idx * 6 + 3];
        r4 = rows_raw[(size_t)idx * 6 + 4];
        r5 = rows_raw[(size_t)idx * 6 + 5];
        if (!(r4 > CONF_T)) { r0 = r1 = r2 = r3 = r4 = r5 = 0.f; }
    }
    // rows: [0, 180000)
    out[(size_t)i * 6 + 0] = r0;
    out[(size_t)i * 6 + 1] = r1;
    out[(size_t)i * 6 + 2] = r2;
    out[(size_t)i * 6 + 3] = r3;
    out[(size_t)i * 6 + 4] = r4;
    out[(size_t)i * 6 + 5] = r5;
    // boxes: [180000, 300000)
    const float cc = r5 * MAX_WH_F;
    float* boxes = out + (size_t)TOPN * 6;
    boxes[(size_t)i * 4 + 0] = r0 + cc;
    boxes[(size_t)i * 4 + 1] = r1 + cc;
    boxes[(size_t)i * 4 + 2] = r2 + cc;
    boxes[(size_t)i * 4 + 3] = r3 + cc;
    // scores: [300000, 330000)
    out[(size_t)TOPN * 10 + i] = r4;
}

extern "C" void kernel_launch(void* const* d_in, const int* in_sizes, int n_in,
                              void* d_out, int out_size, void* d_ws, size_t ws_size,
                              hipStream_t stream)
{
    (void)in_sizes; (void)n_in; (void)out_size; (void)ws_size;
    const float* pred = (const float*)d_in[0];     // (32, 84, 33600); only batch 0 used
    float* out = (float*)d_out;

    // workspace: [keys: 65536 u64 = 512 KB][rows_raw: 33600*6 f32 = 787.5 KB]
    unsigned long long* keys = (unsigned long long*)d_ws;
    float* rows_raw = (float*)((char*)d_ws + (size_t)N_PAD * sizeof(unsigned long long));

    stage1_kernel<<<(N_ANCH + TILE_W - 1) / TILE_W, TILE_W, 0, stream>>>(pred, rows_raw, keys);
    pad_keys_kernel<<<(N_PAD - N_ANCH + 255) / 256, 256, 0, stream>>>(keys);

    for (unsigned k = 2u; k <= (unsigned)N_PAD; k <<= 1) {
        for (unsigned j = k >> 1; j >= 2048u; j >>= 1)
            bitonic_global_kernel<<<(N_PAD / 2) / 256, 256, 0, stream>>>(keys, k, j);
        unsigned jstart = ((k >> 1) < 1024u) ? (k >> 1) : 1024u;
        bitonic_lds_kernel<<<N_PAD / 2048, 1024, 0, stream>>>(keys, k, jstart);
    }

    gather_out_kernel<<<(TOPN + 255) / 256, 256, 0, stream>>>(keys, rows_raw, out);
}